// PBCAgnosticElectrostaticFeatureBlock_53506702574013
// MI455X (gfx1250) — compile-verified
//
#include <hip/hip_runtime.h>

// ---------------------------------------------------------------------------
// PBC-agnostic electrostatic feature block (non-PBC path):
// per-graph all-pairs Coulomb potential V and field F.
//
// Strategy (MI455X / gfx1250, wave32):
//   d2(i,j) = |a_i|^2 + |b_j|^2 - 2*(a_i . b_j),  a_i = p_i + 1e-15, b_j = p_j
//   The Gram term a_i . b_j over a 16x16 tile (K=3 padded to 4) is computed
//   with V_WMMA_F32_16X16X4_F32. Epilogue (rsqrt + 1/d^3 via inv_r^3,
//   q-weighted sums) runs on VALU; j-reduction via 16-lane butterfly +
//   LDS ds_add_f32.   F_i = p_i * (sum_j q_j/d^3) - (sum_j q_j p_j/d^3).
// ---------------------------------------------------------------------------

typedef __attribute__((ext_vector_type(2))) float v2f;
typedef __attribute__((ext_vector_type(8))) float v8f;

#define NUM_GRAPHS 16
#define NPG        1024          // nodes per graph
#define IB         16            // rows per i-block
#define TILES_J    (NPG / 16)    // 64 j-tiles per graph
#define WAVES      8             // waves per block (256 threads, wave32)
#define TPW        (TILES_J / WAVES) // 8 j-tiles per wave
#define FEAT_COLS  18            // NUM_WIDTHS * (1 + 3 + 5)

#if defined(__HIP_DEVICE_COMPILE__)
#if !__has_builtin(__builtin_amdgcn_wmma_f32_16x16x4_f32)
#error "gfx1250 f32 WMMA builtin __builtin_amdgcn_wmma_f32_16x16x4_f32 not available in device pass"
#endif
#endif

__global__ void zero_out_kernel(float* __restrict__ out, int n) {
    int i = blockIdx.x * blockDim.x + threadIdx.x;
    const int stride = gridDim.x * blockDim.x;
    for (; i < n; i += stride) out[i] = 0.0f;
}

__global__ __launch_bounds__(256) void coulomb_wmma_kernel(
    const float* __restrict__ pos,    // (16384, 3) f32
    const float* __restrict__ sfeat,  // (16384, 9, 2) f32 ; q = sfeat[node*18]
    float* __restrict__ out)          // (2, 16384, 18) f32, pre-zeroed
{
    __shared__ float sAx[IB], sAy[IB], sAz[IB], sRa2[IB];
    __shared__ float sRed[IB][5];     // V, S0, Sx, Sy, Sz per row

    const int tid  = threadIdx.x;
    const int lane = tid & 31;
    const int wave = tid >> 5;
    const int half = lane >> 4;       // 0: lanes 0-15, 1: lanes 16-31
    const int lidx = lane & 15;       // N (column) index within tile

    const int bid   = blockIdx.x;
    const int g     = bid >> 6;       // graph id            (16)
    const int i0    = (bid & 63) * IB;// row block in graph  (64)
    const int gbase = g * NPG;

    // ---- stage i-side rows (a_i = p_i + eps) and zero reduction buffer ----
    if (tid < IB * 5) ((float*)sRed)[tid] = 0.0f;
    if (tid < IB) {
        const int node = gbase + i0 + tid;
        const float x = pos[node * 3 + 0] + 1e-15f;
        const float y = pos[node * 3 + 1] + 1e-15f;
        const float z = pos[node * 3 + 2] + 1e-15f;
        sAx[tid] = x; sAy[tid] = y; sAz[tid] = z;
        sRa2[tid] = x * x + y * y + z * z;
    }
    __syncthreads();

    // A operand (16x4 f32, K padded with 0): constant over the whole j loop.
    // lanes 0-15: {K0=x, K1=y}; lanes 16-31: {K2=z, K3=0}
    v2f A;
    A.x = half ? sAz[lidx] : sAx[lidx];
    A.y = half ? 0.0f      : sAy[lidx];

    // |a_m|^2 for the 8 rows this lane accumulates (M = v + 8*half)
    float ra2v[8];
#pragma unroll
    for (int v = 0; v < 8; ++v) ra2v[v] = sRa2[v + 8 * half];

    v8f accV = {}, accS0 = {}, accSx = {}, accSy = {}, accSz = {};

#pragma unroll 2
    for (int t = 0; t < TPW; ++t) {
        const int j0 = (wave * TPW + t) * 16;
        const int jl = j0 + lidx;          // column j in graph (shared by all 8 rows)
        const int jn = gbase + jl;         // global node index
        const float pjx = pos[jn * 3 + 0];
        const float pjy = pos[jn * 3 + 1];
        const float pjz = pos[jn * 3 + 2];
        const float qj  = sfeat[jn * 18];  // source_feats[j, 0, 0]
        const float rb2 = pjx * pjx + pjy * pjy + pjz * pjz;

        // B operand (4x16 f32), mirror of A with N = lidx
        v2f B;
        B.x = half ? pjz : pjx;
        B.y = half ? 0.0f : pjy;

        // Gram tile: G[m][n] = a_m . b_n   (EXEC is all-ones here: no divergence)
        v8f C = {};
        v8f G = __builtin_amdgcn_wmma_f32_16x16x4_f32(
            /*neg_a=*/false, A, /*neg_b=*/false, B,
            /*c_mod=*/(short)0, C, /*reuse_a=*/false, /*reuse_b=*/false);

#pragma unroll
        for (int v = 0; v < 8; ++v) {
            const int m   = v + 8 * half;
            const float d2 = ra2v[v] + rb2 - 2.0f * G[v];   // |r_ij + eps|^2
            const bool diag = (i0 + m) == jl;               // same graph by construction
            const float inv_r = diag ? 0.0f : __frsqrt_rn(d2);
            // inv_r / (d2 + 1e-15)  ==  inv_r^3 to ~1e-15 relative: avoid the
            // IEEE v_div_scale/v_div_fmas sequence (10+ VALU ops per element).
            const float ir3 = inv_r * inv_r * inv_r;        // 0 on diagonal
            const float w3  = qj * ir3;
            accV[v]  += qj * inv_r;
            accS0[v] += w3;
            accSx[v] += w3 * pjx;
            accSy[v] += w3 * pjy;
            accSz[v] += w3 * pjz;
        }
    }

    // ---- reduce over the 16 columns held across each half-wave ----
    // xor masks 1,2,4,8 never flip bit 4, so width-32 shuffles stay in-half.
#pragma unroll
    for (int v = 0; v < 8; ++v) {
        float a0 = accV[v], a1 = accS0[v], a2 = accSx[v], a3 = accSy[v], a4 = accSz[v];
#pragma unroll
        for (int s = 1; s < 16; s <<= 1) {
            a0 += __shfl_xor(a0, s, 32);
            a1 += __shfl_xor(a1, s, 32);
            a2 += __shfl_xor(a2, s, 32);
            a3 += __shfl_xor(a3, s, 32);
            a4 += __shfl_xor(a4, s, 32);
        }
        if (lidx == 0) {                   // one lane per half commits 8 rows
            const int m = v + 8 * half;
            atomicAdd(&sRed[m][0], a0);
            atomicAdd(&sRed[m][1], a1);
            atomicAdd(&sRed[m][2], a2);
            atomicAdd(&sRed[m][3], a3);
            atomicAdd(&sRed[m][4], a4);
        }
    }
    __syncthreads();

    // ---- epilogue: write V and F into both width blocks ----
    if (tid < IB) {
        const int m    = tid;
        const int node = gbase + i0 + m;
        const float V  = sRed[m][0];
        const float S0 = sRed[m][1];
        const float Fx = sAx[m] * S0 - sRed[m][2];
        const float Fy = sAy[m] * S0 - sRed[m][3];
        const float Fz = sAz[m] * S0 - sRed[m][4];
        float* o = out + (size_t)node * FEAT_COLS;
        o[0] = V;  o[1]  = Fx; o[2]  = Fy; o[3]  = Fz;   // width 0
        o[9] = V;  o[10] = Fx; o[11] = Fy; o[12] = Fz;   // width 1
        // cols 4-8, 13-17 and the entire self_terms half stay zero (pre-filled)
    }
}

extern "C" void kernel_launch(void* const* d_in, const int* in_sizes, int n_in,
                              void* d_out, int out_size, void* d_ws, size_t ws_size,
                              hipStream_t stream) {
    // setup_inputs order:
    // 0: k_vectors  1: k_vectors_normed_squared  2: k_vectors_mask
    // 3: source_feats  4: node_positions  5: batch  6: volumes  7: pbc  8: use_pbc_evaluator
    const float* sfeat = (const float*)d_in[3];
    const float* pos   = (const float*)d_in[4];
    float* out = (float*)d_out;

    // Zero everything (covers zero feature columns + the self_terms half).
    int zb = (out_size + 255) / 256;
    if (zb > 2048) zb = 2048;
    zero_out_kernel<<<zb, 256, 0, stream>>>(out, out_size);

    // One block per (graph, 16-row block): 16 * 64 = 1024 blocks, 8 wave32 each.
    coulomb_wmma_kernel<<<NUM_GRAPHS * (NPG / IB), 256, 0, stream>>>(pos, sfeat, out);
}